// DrosophilaOpticLobeCircuit_44203803411110
// MI455X (gfx1250) — compile-verified
//
#include <hip/hip_runtime.h>
#include <stdint.h>
#include <stddef.h>

// ---------------------------------------------------------------------------
// Drosophila optic lobe circuit: 50-step recurrent sparse graph sim.
//  per step:  Tm1 HP->relu->LP filters (25K) ; r = relu(v) ;
//             syn = segment_sum(r[src]*w, tgt) over 6.4M edges ;
//             v += DT * (-v + syn + vrest [+ ext]) / tau   (Tm1 rows frozen)
// Edge arrays (src,tgt,w) = 76.8 MB -> resident in MI455X's 192MB L2 after
// step 0; steady state limited by L2 bandwidth + L2 float-atomic throughput.
// Edge streams are staged into LDS via gfx1250 GLOBAL_LOAD_ASYNC_TO_LDS_B128
// with a double-buffered ASYNCcnt pipeline (confirmed lowering in round 2).
// Neuron-side elementwise kernels are fused (post(t) + pre(t+1)) to halve
// launch count: 102 graph nodes per call instead of 151.
// ---------------------------------------------------------------------------

constexpr int   kNeurons = 200000;
constexpr int   kEdges   = 6400000;
constexpr int   kTm1     = 25000;
constexpr int   kSteps   = 50;
constexpr float kDt      = 0.1f;
constexpr float kTauHp   = 12.3f;
constexpr float kTauLp   = 2.3f;

constexpr int kEdgeTile  = 1024;               // 256 threads x 4 edges
constexpr int kTiles     = kEdges / kEdgeTile; // 6250 exactly

#ifndef __has_builtin
#define __has_builtin(x) 0
#endif

#if defined(__HIP_DEVICE_COMPILE__) && \
    __has_builtin(__builtin_amdgcn_global_load_async_to_lds_b128) && \
    __has_builtin(__builtin_amdgcn_s_wait_asynccnt)
#define USE_ASYNC_LDS 1
#else
#define USE_ASYNC_LDS 0
#endif

// ---------------------------------------------------------------------------
// Device helpers (element-local pieces of the recurrence)
// ---------------------------------------------------------------------------
__device__ __forceinline__ void tm1_advance(int i, const float* __restrict__ x_t,
                                            float* __restrict__ f,
                                            float* __restrict__ tv,
                                            float& vi) {
  float x   = x_t[i];
  float fo  = f[i];
  float tvo = tv[i];
  float hp  = x - fo;                       // high-pass with OLD f
  f[i]  = fo + kDt * (x - fo) / kTauHp;     // f update
  float rect = fmaxf(hp, 0.0f);
  tv[i] = tvo + kDt * (rect - tvo) / kTauLp;
  vi = tvo;                                 // v[:n_tm1] = OLD tm1_v
}

// ---------------------------------------------------------------------------
// Kernel: pre-pass for step 0 only (no prior Euler update to apply).
// ---------------------------------------------------------------------------
__global__ __launch_bounds__(256) void k_pre(
    const float* __restrict__ x_t,   // tm1_input row for step 0 [kTm1]
    float* __restrict__ v,
    float* __restrict__ f,
    float* __restrict__ tv,
    float* __restrict__ r,
    float* __restrict__ syn) {
  int i = blockIdx.x * blockDim.x + threadIdx.x;
  if (i >= kNeurons) return;
  float vi = v[i];
  if (i < kTm1) {
    tm1_advance(i, x_t, f, tv, vi);
    v[i] = vi;
  }
  r[i]   = fmaxf(vi, 0.0f);
  syn[i] = 0.0f;
}

// ---------------------------------------------------------------------------
// Fused kernel: Euler update for step t (Tm1 rows frozen: dv forced to 0 in
// the reference), then pre-pass for step t+1 (Tm1 filters, r=relu(v), syn=0).
// Every piece is element-local, so fusion is exact.
// ---------------------------------------------------------------------------
__global__ __launch_bounds__(256) void k_mid(
    const float* __restrict__ x_next,  // tm1_input row for step t+1
    float* __restrict__ v,
    float* __restrict__ f,
    float* __restrict__ tv,
    float* __restrict__ r,
    float* __restrict__ syn,
    const float* __restrict__ vrest,
    const float* __restrict__ tau) {
  int i = blockIdx.x * blockDim.x + threadIdx.x;
  if (i >= kNeurons) return;
  float vi = v[i];
  if (i >= kTm1) {
    // post(t): v += DT * (-v + syn + vrest) / tau
    vi += kDt * ((-vi + syn[i] + vrest[i]) / tau[i]);
  } else {
    // pre(t+1): advance Tm1 filters, v[:n_tm1] = old tm1_v
    tm1_advance(i, x_next, f, tv, vi);
  }
  v[i]   = vi;
  r[i]   = fmaxf(vi, 0.0f);
  syn[i] = 0.0f;
}

// ---------------------------------------------------------------------------
// Final kernel: Euler update for the last step, write result.
// ---------------------------------------------------------------------------
__global__ __launch_bounds__(256) void k_post(
    float* __restrict__ v,
    const float* __restrict__ syn,
    const float* __restrict__ vrest,
    const float* __restrict__ tau,
    float* __restrict__ out) {
  int i = blockIdx.x * blockDim.x + threadIdx.x;
  if (i >= kNeurons) return;
  float vi = v[i];
  if (i >= kTm1) {
    vi += kDt * ((-vi + syn[i] + vrest[i]) / tau[i]);
    v[i] = vi;
  }
  out[i] = vi;
}

// ---------------------------------------------------------------------------
// Edge kernel: gather r[src] * w, scatter-add into syn.
// Async-DMA path: double-buffered LDS staging of the three linear edge
// streams via GLOBAL_LOAD_ASYNC_TO_LDS_B128; each wave writes/reads only its
// own 32 LDS slots, so ordering is enforced purely by ASYNCcnt (no barriers).
// ---------------------------------------------------------------------------
#if USE_ASYNC_LDS
// Builtin signature (probe-confirmed): (v4i addrspace(1)*, v4i addrspace(3)*,
// imm offset, imm cpol).
typedef int v4i __attribute__((__vector_size__(16)));
typedef __attribute__((address_space(1))) v4i as1_v4i;
typedef __attribute__((address_space(3))) v4i as3_v4i;

#define ASYNC_CP128(gp, lp)                                                    \
  __builtin_amdgcn_global_load_async_to_lds_b128(                              \
      (as1_v4i*)(gp), (as3_v4i*)(lp), 0, 0)
#endif

__global__ __launch_bounds__(256) void k_edges(
    const int4*  __restrict__ src4,
    const int4*  __restrict__ tgt4,
    const float4* __restrict__ w4,
    const float* __restrict__ r,
    float* __restrict__ syn) {
  const int lane = threadIdx.x;
#if USE_ASYNC_LDS
  __shared__ int4   sS[2][256];
  __shared__ int4   sT[2][256];
  __shared__ float4 sW[2][256];

  int tile = blockIdx.x;
  if (tile >= kTiles) return;

  // prologue: issue DMA for first tile into buffer 0 (ASYNCcnt = 3)
  {
    int g = tile * 256 + lane;
    ASYNC_CP128(src4 + g, &sS[0][lane]);
    ASYNC_CP128(tgt4 + g, &sT[0][lane]);
    ASYNC_CP128(w4   + g, &sW[0][lane]);
  }
  int p = 0;
  for (; tile < kTiles; tile += (int)gridDim.x) {
    int nxt = tile + (int)gridDim.x;
    if (nxt < kTiles) {
      // issue next tile into the other buffer, then wait for current tile;
      // async loads complete in order per wave, so asynccnt<=3 after issuing
      // 3 more guarantees the current tile's 3 transfers have landed.
      int g = nxt * 256 + lane;
      ASYNC_CP128(src4 + g, &sS[p ^ 1][lane]);
      ASYNC_CP128(tgt4 + g, &sT[p ^ 1][lane]);
      ASYNC_CP128(w4   + g, &sW[p ^ 1][lane]);
      __builtin_amdgcn_s_wait_asynccnt(3);
    } else {
      __builtin_amdgcn_s_wait_asynccnt(0);   // drain
    }
    int4   s = sS[p][lane];
    int4   t = sT[p][lane];
    float4 w = sW[p][lane];
    float r0 = r[s.x], r1 = r[s.y], r2 = r[s.z], r3 = r[s.w];
    atomicAdd(&syn[t.x], r0 * w.x);
    atomicAdd(&syn[t.y], r1 * w.y);
    atomicAdd(&syn[t.z], r2 * w.z);
    atomicAdd(&syn[t.w], r3 * w.w);
    p ^= 1;
  }
#else
  // fallback: vector b128 loads + explicit prefetch lookahead
  for (int tile = blockIdx.x; tile < kTiles; tile += (int)gridDim.x) {
    int g  = tile * 256 + lane;
    int gn = g + (int)gridDim.x * 256;
    if (gn < kTiles * 256) {
      __builtin_prefetch(src4 + gn, 0, 0);
      __builtin_prefetch(tgt4 + gn, 0, 0);
      __builtin_prefetch(w4   + gn, 0, 0);
    }
    int4   s = src4[g];
    int4   t = tgt4[g];
    float4 w = w4[g];
    float r0 = r[s.x], r1 = r[s.y], r2 = r[s.z], r3 = r[s.w];
    atomicAdd(&syn[t.x], r0 * w.x);
    atomicAdd(&syn[t.y], r1 * w.y);
    atomicAdd(&syn[t.z], r2 * w.z);
    atomicAdd(&syn[t.w], r3 * w.w);
  }
#endif
}

// ---------------------------------------------------------------------------
// State init: must re-zero every call (harness does not re-poison d_ws).
// ---------------------------------------------------------------------------
__global__ __launch_bounds__(256) void k_init(
    float* __restrict__ v, float* __restrict__ f, float* __restrict__ tv) {
  int i = blockIdx.x * blockDim.x + threadIdx.x;
  if (i < kNeurons) v[i] = 0.0f;
  if (i < kTm1) { f[i] = 0.0f; tv[i] = 0.0f; }
}

// ---------------------------------------------------------------------------
extern "C" void kernel_launch(void* const* d_in, const int* in_sizes, int n_in,
                              void* d_out, int out_size, void* d_ws, size_t ws_size,
                              hipStream_t stream) {
  (void)in_sizes; (void)n_in; (void)out_size; (void)ws_size;
  const float* tm1_input = (const float*)d_in[0];  // [50, 25000]
  const float* weights   = (const float*)d_in[1];  // [6.4M]
  const float* tau       = (const float*)d_in[2];  // [200K]
  const float* vrest     = (const float*)d_in[3];  // [200K]
  const int*   src       = (const int*)d_in[4];    // [6.4M]
  const int*   tgt       = (const int*)d_in[5];    // [6.4M]
  // d_in[6] = n_tm1 scalar (compile-time constant 25000 here)
  float* out = (float*)d_out;

  float* ws  = (float*)d_ws;
  float* v   = ws;                 // kNeurons
  float* f   = v  + kNeurons;      // kTm1
  float* tv  = f  + kTm1;          // kTm1
  float* r   = tv + kTm1;          // kNeurons
  float* syn = r  + kNeurons;      // kNeurons   (total ~2.6 MB)

  const int nBlocks = (kNeurons + 255) / 256;
  const int eBlocks = 2048;        // ~3 tiles/block -> pipeline has work

  k_init<<<nBlocks, 256, 0, stream>>>(v, f, tv);
  k_pre<<<nBlocks, 256, 0, stream>>>(tm1_input, v, f, tv, r, syn);

  for (int t = 0; t < kSteps - 1; ++t) {
    k_edges<<<eBlocks, 256, 0, stream>>>((const int4*)src, (const int4*)tgt,
                                         (const float4*)weights, r, syn);
    k_mid<<<nBlocks, 256, 0, stream>>>(tm1_input + (size_t)(t + 1) * kTm1,
                                       v, f, tv, r, syn, vrest, tau);
  }
  k_edges<<<eBlocks, 256, 0, stream>>>((const int4*)src, (const int4*)tgt,
                                       (const float4*)weights, r, syn);
  k_post<<<nBlocks, 256, 0, stream>>>(v, syn, vrest, tau, out);
}